// SNNExtractorSingle_68049461838256
// MI455X (gfx1250) — compile-verified
//
#include <hip/hip_runtime.h>
#include <hip/hip_bf16.h>

// SNN Leaky (reset-by-subtract) over T steps, elementwise on N values.
//   reset_t = (mem_{t-1} > 1) ? 1 : 0        ( == spk_{t-1}, since THR == 1 )
//   mem_t   = 0.95*mem_{t-1} + x - reset_t
//   spk_t   = (mem_t > 1) ? 1 : 0            -> out[t*N + i]
//
// Store-bandwidth bound: 629 MB f32 out vs 12.6 MB in -> ~27.5 us floor
// at 23.3 TB/s. Output > 192 MB L2 and write-once => non-temporal stores.
// 4 elems/thread => global_store_b128 th:TH_STORE_NT per step, 4-way ILP
// inside the serial time recurrence. wave32, 256-thread blocks.

typedef __attribute__((ext_vector_type(4))) float v4f;

#define SNN_BETA 0.95f
#define SNN_THR  1.0f

__global__ __launch_bounds__(256)
void snn_leaky_scan_kernel(const float* __restrict__ x,
                           float* __restrict__ out,
                           int n, int T) {
    // All indices fit in int32: N = 3.1M, T*N = 157M elements (< 2^31),
    // max byte offset 629 MB (< 2^31).
    const int base = (int)(blockIdx.x * blockDim.x + threadIdx.x) * 4;

    if (base + 3 < n) {
        // Fast path: NT b128 load of x, one NT b128 store per time step.
        v4f xv = __builtin_nontemporal_load((const v4f*)(x + base));
        v4f m = {0.0f, 0.0f, 0.0f, 0.0f};
        v4f r = {0.0f, 0.0f, 0.0f, 0.0f};   // reset_0 = (mem0=0 > 1) = 0
        float* op = out + base;

#pragma unroll 5
        for (int t = 0; t < T; ++t) {
            v4f s;
#pragma unroll
            for (int k = 0; k < 4; ++k) {
                m[k] = __builtin_fmaf(SNN_BETA, m[k], xv[k]) - r[k];
                s[k] = (m[k] > SNN_THR) ? 1.0f : 0.0f;
            }
            __builtin_nontemporal_store(s, (v4f*)op);
            r = s;      // spk_t == reset_{t+1} (THR == 1.0)
            op += n;    // next time-step plane (stride N floats = 12.6 MB)
        }
    } else if (base < n) {
        // Scalar tail (unused when N % 4 == 0; kept for generality).
        for (int i = base; i < n; ++i) {
            const float xs = x[i];
            float m = 0.0f, r = 0.0f;
            float* op = out + i;
#pragma unroll 5
            for (int t = 0; t < T; ++t) {
                m = __builtin_fmaf(SNN_BETA, m, xs) - r;
                const float s = (m > SNN_THR) ? 1.0f : 0.0f;
                __builtin_nontemporal_store(s, op);
                r = s;
                op += n;
            }
        }
    }
}

extern "C" void kernel_launch(void* const* d_in, const int* in_sizes, int n_in,
                              void* d_out, int out_size, void* d_ws, size_t ws_size,
                              hipStream_t stream) {
    (void)n_in; (void)d_ws; (void)ws_size;

    const float* x = (const float*)d_in[0];        // [1024,3,32,32] f32
    float* out     = (float*)d_out;                // [T*N] f32, t-major

    const int n = in_sizes[0];                     // 3,145,728
    const int T = (n > 0) ? (out_size / n) : 0;    // 50 (derived host-side;
                                                   //  scalar inputs live on device)

    const int nvec    = (n + 3) / 4;
    const int threads = 256;                       // 8 wave32 waves per block
    const int blocks  = (nvec + threads - 1) / threads;

    snn_leaky_scan_kernel<<<blocks, threads, 0, stream>>>(x, out, n, T);
}